// ConvNAT_6914897346958
// MI455X (gfx1250) — compile-verified
//
#include <hip/hip_runtime.h>
#include <hip/hip_bf16.h>
#include <math.h>

typedef __attribute__((ext_vector_type(2))) float v2f;
typedef __attribute__((ext_vector_type(8))) float v8f;

#define HDIM 96
#define WDIM 96
#define BSZ  2
#define CIN  64
#define HW   (HDIM * WDIM)      /* 9216  */
#define MTOT (BSZ * HW)         /* 18432 */
#define CCAT 128                /* q/k after PE concat; also Cv */
#define NA_K 7
#define NA_D 2
/* Lg = 96/2 = 48, max clamp start = Lg - K = 41 */
#define NA_MAXSTART 41

/* LDS patch: [CIN][3 rows][18 cols] floats = 13824 B */
#define PATCH_COLS 18
#define PATCH_ELEMS (CIN * 3 * PATCH_COLS)

// ---------------------------------------------------------------------------
// Fused QKV conv3x3 (pad 1) as implicit GEMM on V_WMMA_F32_16X16X4_F32.
// One 512-thread block (16 waves) per 16-position M-tile (one image row
// segment; 96%16==0 so tiles never cross rows/batches). The block stages the
// 3x18x64 input patch (zero-filled halo) into LDS once with coalesced loads;
// each wave then computes one 16x16 output tile: waves 0-3 = q, 4-7 = k,
// 8-15 = v. K = Cin*9 = 576, stepped 4 per WMMA -> 144 WMMA per wave.
// Outputs stored channels-last: qc/kc get conv in ch 0..63 (PE fills 64..127),
// v gets all 128 channels.
// ---------------------------------------------------------------------------
__global__ __launch_bounds__(512)
void conv_qkv_wmma(const float* __restrict__ x,
                   const float* __restrict__ wq, const float* __restrict__ bq,
                   const float* __restrict__ wk, const float* __restrict__ bk,
                   const float* __restrict__ wv, const float* __restrict__ bv,
                   float* __restrict__ qc, float* __restrict__ kc,
                   float* __restrict__ vv)
{
    __shared__ float patch[PATCH_ELEMS];   // [c][row3][col18]

    const int tid    = threadIdx.x;
    const int lane   = tid & 31;
    const int wid    = tid >> 5;           // 0..15
    const int tile_m = blockIdx.x;

    // Tile geometry (16 consecutive flat positions = one row segment)
    const int m0 = tile_m * 16;
    const int b  = m0 / HW;
    const int hw = m0 - b * HW;
    const int h  = hw / WDIM;
    const int w0 = hw - h * WDIM;          // multiple of 16

    // ---- Stage input patch (coalesced along w), zero-fill halo ------------
    const float* xb = x + (size_t)b * CIN * HW;
    for (int e = tid; e < PATCH_ELEMS; e += 512) {
        const int c   = e / (3 * PATCH_COLS);
        const int rem = e - c * (3 * PATCH_COLS);
        const int r   = rem / PATCH_COLS;
        const int col = rem - r * PATCH_COLS;
        const int ih  = h + r - 1;
        const int iw  = w0 + col - 1;
        float v = 0.0f;
        if (ih >= 0 && ih < HDIM && iw >= 0 && iw < WDIM)
            v = xb[(size_t)c * HW + (size_t)ih * WDIM + iw];
        patch[e] = v;
    }
    __syncthreads();

    // ---- Per-wave GEMM tile ----------------------------------------------
    const float* wsel; const float* bsel; float* osel; int ntl;
    if (wid < 4)      { wsel = wq; bsel = bq; osel = qc; ntl = wid;     }
    else if (wid < 8) { wsel = wk; bsel = bk; osel = kc; ntl = wid - 4; }
    else              { wsel = wv; bsel = bv; osel = vv; ntl = wid - 8; }

    // A fragment (16x4 f32): lanes 0-15 M=lane with K pair {0,1};
    // lanes 16-31 same M range with K pair {2,3}. VGPR0=K even, VGPR1=K odd.
    const int mA   = lane & 15;
    const int kprA = (lane >> 4) << 1;          // 0 or 2
    // B fragment (4x16 f32): lane&15 = N column, half-wave selects K pair.
    const int nB   = lane & 15;
    const int kprB = (lane >> 4) << 1;
    const int co   = ntl * 16 + nB;             // output channel (local)

    __builtin_prefetch(wsel + (size_t)co * CIN * 9, 0, 1);  // global_prefetch_b8

    v8f acc = {};
    for (int dy = 0; dy < 3; ++dy) {
        for (int dx = 0; dx < 3; ++dx) {
            const float* pp = &patch[(size_t)dy * PATCH_COLS + mA + dx];
            const float* wp = wsel + (size_t)co * CIN * 9 + dy * 3 + dx;
            for (int c0 = 0; c0 < CIN; c0 += 4) {
                v2f a, bf;
                const int ca = c0 + kprA;
                a.x = pp[(size_t)(ca    ) * (3 * PATCH_COLS)];
                a.y = pp[(size_t)(ca + 1) * (3 * PATCH_COLS)];
                const int cb = c0 + kprB;
                bf.x = wp[(size_t)(cb    ) * 9];
                bf.y = wp[(size_t)(cb + 1) * 9];
                // (neg_a, A, neg_b, B, c_mod, C, reuse_a, reuse_b)
                acc = __builtin_amdgcn_wmma_f32_16x16x4_f32(
                        false, a, false, bf, (short)0, acc, false, false);
            }
        }
    }

    // C/D layout: VGPR g -> row (g + (lane>>4)*8), col = lane&15.
    const float bias      = bsel[co];
    const int   mrow_base = tile_m * 16 + ((lane >> 4) << 3);
#pragma unroll
    for (int g = 0; g < 8; ++g) {
        const int mrow = mrow_base + g;
        osel[(size_t)mrow * CCAT + ntl * 16 + nB] = acc[g] + bias;
    }
}

// ---------------------------------------------------------------------------
// Write sinusoidal PE into channels 64..127 of q_cat and k_cat.
// pe channel c<32 encodes H (sincos_1d(96,32)), c>=32 encodes W.
// ---------------------------------------------------------------------------
__global__ void pe_fill(float* __restrict__ qc, float* __restrict__ kc)
{
    const int idx = blockIdx.x * blockDim.x + threadIdx.x;
    if (idx >= MTOT * 64) return;
    const int c  = idx & 63;
    const int m  = idx >> 6;
    const int hw = m % HW;
    const int h  = hw / WDIM;
    const int w  = hw - h * WDIM;

    int pos, cc;
    if (c < 32) { pos = h; cc = c; } else { pos = w; cc = c - 32; }
    const int   j    = (cc < 16) ? cc : cc - 16;
    // inv_freq[j] = 10000^(-2j/32) = exp(-j * ln(10000)/16)
    const float invf = expf(-(float)j * (9.210340371976184f / 16.0f));
    const float ang  = (float)pos * invf;
    const float val  = (cc < 16) ? sinf(ang) : cosf(ang);

    qc[(size_t)m * CCAT + 64 + c] = val;
    kc[(size_t)m * CCAT + 64 + c] = val;
}

// ---------------------------------------------------------------------------
// Dilated 7x7 neighborhood attention, clamped borders (NATTEN semantics).
// One 128-thread block per query position. Threads 0..48 compute logits
// (float4-vectorized dot), LDS softmax, then thread t owns output channel t
// (coalesced v reads, channels-last).
// ---------------------------------------------------------------------------
__global__ __launch_bounds__(128)
void na2d_attn(const float* __restrict__ qc, const float* __restrict__ kc,
               const float* __restrict__ vv, float* __restrict__ out)
{
    const int m  = blockIdx.x;
    const int b  = m / HW;
    const int hw = m - b * HW;
    const int h  = hw / WDIM;
    const int w  = hw - h * WDIM;
    const int t  = threadIdx.x;

    __shared__ float lg[49];
    __shared__ float ex[49];

    const int hg = h & 1, hr = h >> 1;
    int hstart = hr - (NA_K / 2);
    hstart = hstart < 0 ? 0 : (hstart > NA_MAXSTART ? NA_MAXSTART : hstart);
    const int wg = w & 1, wr = w >> 1;
    int wstart = wr - (NA_K / 2);
    wstart = wstart < 0 ? 0 : (wstart > NA_MAXSTART ? NA_MAXSTART : wstart);

    if (t < 49) {
        const int i  = t / 7;
        const int j  = t - i * 7;
        const int hi = hg + (hstart + i) * NA_D;
        const int wj = wg + (wstart + j) * NA_D;
        const float4* qp = (const float4*)(qc + (size_t)m * CCAT);
        const float4* kp = (const float4*)(kc +
                ((size_t)b * HW + (size_t)hi * WDIM + wj) * CCAT);
        float s = 0.0f;
#pragma unroll 8
        for (int c = 0; c < CCAT / 4; ++c) {
            const float4 a = qp[c];
            const float4 k4 = kp[c];
            s += a.x * k4.x + a.y * k4.y + a.z * k4.z + a.w * k4.w;
        }
        lg[t] = s * 0.08838834764831845f;   // 128^-0.5
    }
    __syncthreads();

    float mx = lg[0];
    for (int u = 1; u < 49; ++u) mx = fmaxf(mx, lg[u]);
    if (t < 49) ex[t] = __expf(lg[t] - mx);
    __syncthreads();

    float sum = 0.0f;
    for (int u = 0; u < 49; ++u) sum += ex[u];
    const float inv = 1.0f / sum;

    float accv = 0.0f;
    for (int u = 0; u < 49; ++u) {
        const int i  = u / 7;
        const int j  = u - i * 7;
        const int hi = hg + (hstart + i) * NA_D;
        const int wj = wg + (wstart + j) * NA_D;
        accv += ex[u] * vv[((size_t)b * HW + (size_t)hi * WDIM + wj) * CCAT + t];
    }
    out[(size_t)m * CCAT + t] = accv * inv;
}

// ---------------------------------------------------------------------------
extern "C" void kernel_launch(void* const* d_in, const int* in_sizes, int n_in,
                              void* d_out, int out_size, void* d_ws, size_t ws_size,
                              hipStream_t stream)
{
    const float* x  = (const float*)d_in[0];
    const float* wq = (const float*)d_in[1];
    const float* bq = (const float*)d_in[2];
    const float* wk = (const float*)d_in[3];
    const float* bk = (const float*)d_in[4];
    const float* wv = (const float*)d_in[5];
    const float* bv = (const float*)d_in[6];

    float* qc = (float*)d_ws;                         // (MTOT,128)
    float* kc = qc + (size_t)MTOT * CCAT;             // (MTOT,128)
    float* vv = kc + (size_t)MTOT * CCAT;             // (MTOT,128)
    float* out = (float*)d_out;                       // (MTOT,128)

    conv_qkv_wmma<<<MTOT / 16, 512, 0, stream>>>(x, wq, bq, wk, bk, wv, bv,
                                                 qc, kc, vv);

    const int npe = MTOT * 64;
    pe_fill<<<(npe + 255) / 256, 256, 0, stream>>>(qc, kc);

    na2d_attn<<<MTOT, 128, 0, stream>>>(qc, kc, vv, out);
}